// SuperpointGenerator_82781199663404
// MI455X (gfx1250) — compile-verified
//
#include <hip/hip_runtime.h>
#include <stdint.h>

// ---------------------------------------------------------------------------
// SuperpointGenerator for MI455X (gfx1250).
// Bandwidth-bound voxel-hash / top-k-label pipeline (~190 MB traffic -> ~8us
// floor at 23.3 TB/s; no matrix math => no WMMA by design). CDNA5 paths used:
//   - TENSOR_LOAD_TO_LDS (TDM) bulk-stages the 96 MB coord stream, one
//     descriptor per 3KB tile, double buffered with s_wait_tensorcnt,
//   - per-lane GLOBAL_LOAD_ASYNC_TO_LDS_B128 (+ s_wait_asynccnt) stages the
//     vid stream in the relabel pass,
//   - nontemporal stores for the write-once output keep the hash tables
//     L2-resident (192 MB L2),
//   - global_prefetch on the sequential table scan.
// ---------------------------------------------------------------------------

namespace {
constexpr int   kB         = 16;
constexpr int   kN         = 500000;
constexpr int   kMaxSP     = 512;
constexpr int   kTblBits   = 18;
constexpr int   kTbl       = 1 << kTblBits;      // 262144 slots / batch
constexpr int   kEmpty     = (int)0x80000000;    // impossible vid
constexpr int   kTilePts   = 256;                // points per LDS tile
constexpr int   kTileElems = kTilePts * 3;       // 768 floats = 3072 B
constexpr int   kNBin      = 4096;               // count-histogram bins
constexpr int   kCap       = 2048;               // gather capacity
}

typedef unsigned int u32x4 __attribute__((ext_vector_type(4)));
typedef unsigned int u32x8 __attribute__((ext_vector_type(8)));

__device__ __forceinline__ unsigned hash_vid(int v) {
    return ((unsigned)v * 2654435761u) >> (32 - kTblBits);
}

// ---------------------------------------------------------------------------
// 0) re-init hash tables each launch (harness does not re-poison between runs)
// ---------------------------------------------------------------------------
__global__ __launch_bounds__(256) void sp_init_kernel(int* __restrict__ keys,
                                                      int* __restrict__ cnts,
                                                      int* __restrict__ labels,
                                                      int n) {
    int i = blockIdx.x * 256 + threadIdx.x;
    if (i < n) {
        keys[i]   = kEmpty;
        cnts[i]   = 0;
        labels[i] = -1;
    }
}

// ---------------------------------------------------------------------------
// 1) stream coords through LDS via the Tensor Data Mover (double buffered),
//    compute vids, count points per voxel in an open-addressed hash table.
// ---------------------------------------------------------------------------
__global__ __launch_bounds__(256) void sp_count_kernel(const float* __restrict__ coords,
                                                       int* __restrict__ keys,
                                                       int* __restrict__ cnts,
                                                       int* __restrict__ vids) {
    __shared__ float lbuf[2][kTileElems];

    const int b = blockIdx.y;
    const int t = threadIdx.x;
    const unsigned long long gbase =
        (unsigned long long)(const void*)(coords + (size_t)b * kN * 3);
    const int batchElems = kN * 3;
    const int nTiles     = (kN + kTilePts - 1) / kTilePts;

    int* K = keys + (size_t)b * kTbl;
    int* C = cnts + (size_t)b * kTbl;

    // wave-uniform scalar predicate: only wave 0 drives the TDM
    // (tensor ops ignore EXEC, so we need a *scalar* branch, not lane masking)
    const bool isWave0 = (__builtin_amdgcn_readfirstlane((int)threadIdx.x) < 32);

    // Build a 1-D TDM descriptor for one tile and issue TENSOR_LOAD_TO_LDS.
    auto issue = [&](int tile, int bufIdx) {
        const int startElem = tile * kTileElems;
        int remain = batchElems - startElem;
        const int tileElems = remain < kTileElems ? remain : kTileElems;
        const unsigned long long ga = gbase + (unsigned long long)startElem * 4ull;
        const unsigned lds = (unsigned)(size_t)(void*)&lbuf[bufIdx][0];

        u32x4 g0;   // D# group 0: flags | lds_addr | global_addr | type
        g0[0] = 1u;                                              // count=1, user D#
        g0[1] = lds;                                             // lds_addr [63:32]
        g0[2] = (unsigned)ga;                                    // global_addr[31:0]
        g0[3] = (unsigned)((ga >> 32) & 0x01FFFFFFull) | (2u << 30); // addr[56:32], type=2

        u32x8 g1;   // D# group 1: sizes/dims (1-D tile, 4-byte elements)
        g1[0] = (2u << 16);                                      // data_size=4B, wg_mask=0
        g1[1] = ((unsigned)remain & 0xFFFFu) << 16;              // tensor_dim0[15:0] @ [63:48]
        g1[2] = ((unsigned)remain >> 16) | (1u << 16);           // tensor_dim0[31:16] | tensor_dim1=1
        g1[3] = ((unsigned)tileElems << 16);                     // tensor_dim1 hi=0 | tile_dim0
        g1[4] = 1u;                                              // tile_dim1=1, tile_dim2=0
        g1[5] = (unsigned)remain;                                // tensor_dim0_stride lo (benign)
        g1[6] = 0u;
        g1[7] = 0u;

        asm volatile("tensor_load_to_lds %0, %1" :: "s"(g0), "s"(g1) : "memory");
    };

    const int first = blockIdx.x;
    if (isWave0 && first < nTiles) issue(first, 0);
    int buf = 0;

    for (int tile = first; tile < nTiles; tile += gridDim.x) {
        int next = tile + gridDim.x;
        if (isWave0) {
            if (next < nTiles) {
                issue(next, buf ^ 1);
                __builtin_amdgcn_s_wait_tensorcnt(1);   // current tile landed
            } else {
                __builtin_amdgcn_s_wait_tensorcnt(0);
            }
        }
        __syncthreads();                                // LDS visible block-wide

        int p = tile * kTilePts + t;
        if (p < kN) {
            const float* lp = &lbuf[buf][t * 3];
            // faithful to reference: divide by 0.2f, trunc toward zero
            int vx = (int)(lp[0] / 0.2f);
            int vy = (int)(lp[1] / 0.2f);
            int vz = (int)(lp[2] / 0.2f);
            int vid = vx * 10000 + vy * 100 + vz;
            vids[(size_t)b * kN + p] = vid;

            unsigned h = hash_vid(vid);
            for (int probe = 0; probe < kTbl; ++probe) {
                int k = K[h];
                if (k == kEmpty) {
                    k = atomicCAS(&K[h], kEmpty, vid);
                    if (k == kEmpty) k = vid;           // we claimed the slot
                }
                if (k == vid) { atomicAdd(&C[h], 1); break; }
                h = (h + 1) & (kTbl - 1);
            }
        }
        __syncthreads();                                // protect buf reuse
        buf ^= 1;
    }
}

// ---------------------------------------------------------------------------
// 2) per-batch top-512 selection: LDS count-histogram -> threshold -> gather
//    candidates -> exact rank by (count desc, vid asc). vid-ascending order
//    equals the reference's sorted-unique group-id tiebreak. If <=512 uniques,
//    rank by vid ascending (== inverse indices).
// ---------------------------------------------------------------------------
__global__ __launch_bounds__(1024) void sp_select_kernel(const int* __restrict__ keys,
                                                         const int* __restrict__ cnts,
                                                         int* __restrict__ labels) {
    __shared__ int hist[kNBin];
    __shared__ int svid[kCap];
    __shared__ int scnt[kCap];
    __shared__ int sslot[kCap];
    __shared__ int sCursor, sUniq, sThresh;

    const int b = blockIdx.x;
    const int* K = keys + (size_t)b * kTbl;
    const int* C = cnts + (size_t)b * kTbl;
    int* L = labels + (size_t)b * kTbl;

    for (int i = threadIdx.x; i < kNBin; i += 1024) hist[i] = 0;
    if (threadIdx.x == 0) { sCursor = 0; sUniq = 0; sThresh = 1; }
    __syncthreads();

    int myUniq = 0;
    for (int i = threadIdx.x; i < kTbl; i += 1024) {
        if (i + 4096 < kTbl) __builtin_prefetch(&K[i + 4096], 0, 0);
        if (K[i] != kEmpty) {
            ++myUniq;
            int c = C[i];
            atomicAdd(&hist[c < kNBin ? c : (kNBin - 1)], 1);
        }
    }
    atomicAdd(&sUniq, myUniq);
    __syncthreads();

    if (threadIdx.x == 0) {
        int s = 0, T = 1;
        for (int c = kNBin - 1; c >= 1; --c) {
            s += hist[c];
            if (s >= kMaxSP) { T = c; break; }
            T = 1;
        }
        sThresh = T;
    }
    __syncthreads();

    const int T = sThresh;
    for (int i = threadIdx.x; i < kTbl; i += 1024) {
        if (K[i] != kEmpty && C[i] >= T) {
            int idx = atomicAdd(&sCursor, 1);
            if (idx < kCap) { svid[idx] = K[i]; scnt[idx] = C[i]; sslot[idx] = i; }
        }
    }
    __syncthreads();

    const int M = (sCursor < kCap) ? sCursor : kCap;
    const bool small = (sUniq <= kMaxSP);

    for (int j = threadIdx.x; j < M; j += 1024) {
        int cj = scnt[j], vj = svid[j];
        int rank = 0;
        for (int k = 0; k < M; ++k) {
            int ck = scnt[k], vk = svid[k];
            bool better = small ? (vk < vj)
                                : (ck > cj || (ck == cj && vk < vj));
            rank += better ? 1 : 0;
        }
        if (small || rank < kMaxSP) L[sslot[j]] = rank;
    }
}

// ---------------------------------------------------------------------------
// 3) relabel every point: stream vids through LDS with per-lane async copies
//    (ASYNCcnt path), probe the L2-resident table, NT-store int32 labels.
// ---------------------------------------------------------------------------
__global__ __launch_bounds__(256) void sp_label_kernel(const int* __restrict__ vids,
                                                       const int* __restrict__ keys,
                                                       const int* __restrict__ labels,
                                                       int* __restrict__ out) {
    __shared__ int vtile[2][kTilePts];   // 1 KB per buffer

    const int b = blockIdx.y;
    const int t = threadIdx.x;
    const unsigned long long gbase =
        (unsigned long long)(const void*)(vids + (size_t)b * kN);
    const int batchBytes = kN * 4;
    const int nTiles     = (kN + kTilePts - 1) / kTilePts;

    const int* K = keys   + (size_t)b * kTbl;
    const int* L = labels + (size_t)b * kTbl;

    auto issue = [&](int tile, int bufIdx) {
        int tileByte = tile * (kTilePts * 4);
        int rem = batchBytes - tileByte;
        if (rem > kTilePts * 4) rem = kTilePts * 4;
        int nC16 = rem >> 4;                           // 16B chunks (kN*4 % 16 == 0)
        unsigned lbase = (unsigned)(size_t)(void*)&vtile[bufIdx][0];
        if (t < nC16) {
            unsigned long long g = gbase + (unsigned long long)(tileByte + t * 16);
            unsigned l = lbase + (unsigned)(t * 16);
            asm volatile("global_load_async_to_lds_b128 %0, %1, off"
                         :: "v"(l), "v"(g) : "memory");
        }
    };

    const int first = blockIdx.x;
    if (first < nTiles) issue(first, 0);
    int buf = 0;

    for (int tile = first; tile < nTiles; tile += gridDim.x) {
        int next = tile + gridDim.x;
        if (next < nTiles) {
            issue(next, buf ^ 1);
            asm volatile("s_wait_asynccnt 1" ::: "memory");
        } else {
            asm volatile("s_wait_asynccnt 0" ::: "memory");
        }
        __syncthreads();

        int p = tile * kTilePts + t;
        if (p < kN) {
            const int v = vtile[buf][t];
            unsigned h = hash_vid(v);
            int lab = -1;
            for (int probe = 0; probe < kTbl; ++probe) {
                int k = K[h];
                if (k == v)      { lab = L[h]; break; }
                if (k == kEmpty) break;
                h = (h + 1) & (kTbl - 1);
            }
            // write-once output: NT store keeps hash tables resident in L2
            __builtin_nontemporal_store(lab, &out[(size_t)b * kN + p]);
        }
        __syncthreads();
        buf ^= 1;
    }
}

// ---------------------------------------------------------------------------
extern "C" void kernel_launch(void* const* d_in, const int* in_sizes, int n_in,
                              void* d_out, int out_size, void* d_ws, size_t ws_size,
                              hipStream_t stream) {
    (void)in_sizes; (void)n_in; (void)out_size; (void)ws_size;

    const float* coords = (const float*)d_in[0];
    int* out = (int*)d_out;   // int32 labels (bit-exact in the 4-byte slots)

    // workspace layout
    int* keys   = (int*)d_ws;                 // [kB * kTbl]
    int* cnts   = keys   + (size_t)kB * kTbl; // [kB * kTbl]
    int* labels = cnts   + (size_t)kB * kTbl; // [kB * kTbl]
    int* vids   = labels + (size_t)kB * kTbl; // [kB * kN]

    const int tblTotal = kB * kTbl;
    sp_init_kernel<<<dim3((tblTotal + 255) / 256), dim3(256), 0, stream>>>(
        keys, cnts, labels, tblTotal);

    sp_count_kernel<<<dim3(128, kB), dim3(256), 0, stream>>>(
        coords, keys, cnts, vids);

    sp_select_kernel<<<dim3(kB), dim3(1024), 0, stream>>>(keys, cnts, labels);

    sp_label_kernel<<<dim3(128, kB), dim3(256), 0, stream>>>(
        vids, keys, labels, out);
}